// Model_13898514170164
// MI455X (gfx1250) — compile-verified
//
#include <hip/hip_runtime.h>
#include <math.h>

// CDNA5 / gfx1250. Fused ST-GCN attention block pipeline, fp32 WMMA
// (V_WMMA_F32_16X16X4_F32). Memory-bound (~56 GF vs ~1.5 GB @ 23.3 TB/s),
// so fp32 matrix path is free vs bf16. Inner K-loops are branch-free:
// edge tiles clamp base addresses and guard only the stores.

typedef __attribute__((ext_vector_type(2))) float v2f;
typedef __attribute__((ext_vector_type(8))) float v8f;

#define TDIM 64
#define VDIM 204
#define TV   13056   // TDIM*VDIM

__device__ __forceinline__ v8f wmma4(v2f a, v2f b, v8f c) {
  // (neg_a, A, neg_b, B, c_mod, C, reuse_a, reuse_b)
  return __builtin_amdgcn_wmma_f32_16x16x4_f32(false, a, false, b, (short)0, c,
                                               false, false);
}

// ---------------------------------------------------------------------------
// Generic conv1x1 (channel GEMM) over all (n, t, v) positions.
// MODE 0: out = W*in + b                               (gcn pre-mix)
// MODE 1: out = W*(in + pe) + b                        (qkv projection)
// MODE 2: out = leaky_relu(bn(W*in + b) + res, 0.1)    (on / ff layers)
// No edge tiles: TV = 816*16, Cout multiple of 16.
// ---------------------------------------------------------------------------
template <int MODE, int CIN>
__global__ void __launch_bounds__(256)
conv1x1_wmma(const float* __restrict__ in, const float* __restrict__ pe,
             const float* __restrict__ w, const float* __restrict__ bias,
             const float* __restrict__ bng, const float* __restrict__ bnb,
             const float* __restrict__ bnm, const float* __restrict__ bnv,
             const float* __restrict__ res, float* __restrict__ out,
             int Cout, int nTiles)
{
  int wid = blockIdx.x * 8 + (threadIdx.x >> 5);
  if (wid >= nTiles) return;
  int lane = threadIdx.x & 31;
  int lcol = lane & 15, lhi = lane >> 4;

  int oTiles = Cout >> 4;
  int tilesPerN = oTiles * (TV / 16);
  int n  = wid / tilesPerN;
  int r  = wid - n * tilesPerN;
  int o0 = (r / (TV / 16)) << 4;
  int p0 = (r % (TV / 16)) << 4;

  int p = p0 + lcol;
  const float* wr  = w + (size_t)(o0 + lcol) * CIN + 2 * lhi;   // A row
  const float* inp = in + (size_t)n * CIN * TV + (size_t)(2 * lhi) * TV + p;
  const float* pep = pe + (size_t)(2 * lhi) * TV + p;

  v8f acc = {};
#pragma unroll 4
  for (int kb = 0; kb < CIN; kb += 4) {
    v2f a, b;
    a.x = wr[kb];
    a.y = wr[kb + 1];
    float b0 = inp[(size_t)kb * TV];
    float b1 = inp[(size_t)kb * TV + TV];
    if (MODE == 1) {
      b0 += pep[(size_t)kb * TV];
      b1 += pep[(size_t)kb * TV + TV];
    }
    b.x = b0; b.y = b1;
    acc = wmma4(a, b, acc);
  }
#pragma unroll
  for (int rr = 0; rr < 8; ++rr) {
    int o = o0 + rr + 8 * lhi;
    float val = acc[rr] + bias[o];
    if (MODE >= 2) {
      float s = bng[o] * rsqrtf(bnv[o] + 1e-5f);
      val = val * s + (bnb[o] - bnm[o] * s);
      val += res[((size_t)n * Cout + o) * TV + p];
      val = val > 0.f ? val : 0.1f * val;
    }
    out[((size_t)n * Cout + o) * TV + p] = val;
  }
}

// ---------------------------------------------------------------------------
// Graph mix: OUT[row, u] = sum_v h[row, v] * A[u, v], rows = (n,c,t) flat.
// Fused epilogue: g = relu(bn(OUT) + x).  16-row tiles share one channel c.
// Edge columns (u >= 204): clamped loads compute garbage, stores guarded.
// ---------------------------------------------------------------------------
__global__ void __launch_bounds__(256)
amix_bn_relu_wmma(const float* __restrict__ h, const float* __restrict__ Amat,
                  const float* __restrict__ x,
                  const float* __restrict__ bng, const float* __restrict__ bnb,
                  const float* __restrict__ bnm, const float* __restrict__ bnv,
                  float* __restrict__ gout, int nTiles)
{
  int wid = blockIdx.x * 8 + (threadIdx.x >> 5);
  if (wid >= nTiles) return;
  int lane = threadIdx.x & 31;
  int lcol = lane & 15, lhi = lane >> 4;

  int rt = wid / 13, ct = wid - rt * 13;
  int r0 = rt << 4, u0 = ct << 4;
  int u = u0 + lcol;
  bool uok = (u < VDIM);

  const float* hrow = h + (size_t)(r0 + lcol) * VDIM + 2 * lhi;
  const float* arow = Amat + (size_t)(uok ? u : 0) * VDIM + 2 * lhi;

  v8f acc = {};
#pragma unroll 4
  for (int kb = 0; kb < VDIM; kb += 4) {
    v2f a, b;
    a.x = hrow[kb];
    a.y = hrow[kb + 1];
    b.x = arow[kb];
    b.y = arow[kb + 1];
    acc = wmma4(a, b, acc);
  }
  int c = (r0 >> 6) & 63;                 // row = ((n*64 + c)*64 + t)
  float s   = bng[c] * rsqrtf(bnv[c] + 1e-5f);
  float off = bnb[c] - bnm[c] * s;
#pragma unroll
  for (int rr = 0; rr < 8; ++rr) {
    int row = r0 + rr + 8 * lhi;
    if (uok) {
      float val = acc[rr] * s + off + x[(size_t)row * VDIM + u];
      gout[(size_t)row * VDIM + u] = fmaxf(val, 0.f);
    }
  }
}

// ---------------------------------------------------------------------------
// Attention scores: S[t,q] = sum_{c<16, v<204} Q[c,t,v] K[c,q,v]  per (n,h),
// att = tanh(S/3264)*alpha + att1s.
// Reduction mapped (c outer, v inner) so each lane's two elements are
// contiguous in memory -> b64-mergeable loads.
// ---------------------------------------------------------------------------
__global__ void __launch_bounds__(256)
attn_scores_wmma(const float* __restrict__ qk, const float* __restrict__ alphas,
                 const float* __restrict__ att1s, float* __restrict__ att,
                 int nTiles)
{
  int wid = blockIdx.x * 8 + (threadIdx.x >> 5);
  if (wid >= nTiles) return;
  int lane = threadIdx.x & 31;
  int lcol = lane & 15, lhi = lane >> 4;

  int nh = wid >> 4;               // 16 tiles per (n,h)
  int tl = wid & 15;
  int n  = nh / 3, hh = nh - 3 * n;
  int t0 = (tl >> 2) << 4;
  int q0 = (tl & 3) << 4;

  // Q[c][t][v] = qr[c*TV + v] for row t0+lcol; same for K at row q0+lcol.
  const float* qr = qk + (((size_t)n * 96 + hh * 16) * TDIM + t0 + lcol) * VDIM
                  + 2 * lhi;
  const float* kr = qk + (((size_t)n * 96 + 48 + hh * 16) * TDIM + q0 + lcol) * VDIM
                  + 2 * lhi;

  v8f acc = {};
  for (int cc = 0; cc < 16; ++cc) {
    const float* qc = qr + (size_t)cc * TV;
    const float* kc = kr + (size_t)cc * TV;
#pragma unroll 4
    for (int vb = 0; vb < VDIM; vb += 4) {
      v2f a, b;
      a.x = qc[vb];
      a.y = qc[vb + 1];
      b.x = kc[vb];
      b.y = kc[vb + 1];
      acc = wmma4(a, b, acc);
    }
  }
  float alpha = alphas[hh];
#pragma unroll
  for (int rr = 0; rr < 8; ++rr) {
    int t = t0 + rr + 8 * lhi;
    int q = q0 + lcol;
    float val = tanhf(acc[rr] * (1.f / 3264.f)) * alpha
              + att1s[((size_t)hh * TDIM + t) * TDIM + q];
    att[(((size_t)n * 3 + hh) * TDIM + t) * TDIM + q] = val;
  }
}

// ---------------------------------------------------------------------------
// Apply attention: y2[q,v] = sum_t att[t,q] * g[t,v]  per (n, head s, chan c).
// Edge columns (v >= 204): clamped loads, guarded stores.
// ---------------------------------------------------------------------------
__global__ void __launch_bounds__(256)
attn_apply_wmma(const float* __restrict__ att, const float* __restrict__ g,
                float* __restrict__ y2, int nTiles)
{
  int wid = blockIdx.x * 8 + (threadIdx.x >> 5);
  if (wid >= nTiles) return;
  int lane = threadIdx.x & 31;
  int lcol = lane & 15, lhi = lane >> 4;

  int nsc = wid / 52;              // 4 q-tiles * 13 v-tiles
  int rem = wid - nsc * 52;
  int qt = rem / 13, vt = rem - qt * 13;
  int c = nsc & 63;
  int s = (nsc >> 6) % 3;
  int n = nsc / 192;
  int q0 = qt << 4, v0 = vt << 4;
  int v = v0 + lcol;
  bool vok = (v < VDIM);
  int vc = vok ? v : (VDIM - 1);   // clamped load column

  const float* attb = att + ((size_t)n * 3 + s) * TDIM * TDIM
                    + (size_t)(2 * lhi) * TDIM + q0 + lcol;      // [t][q]
  const float* gb   = g + ((size_t)n * 64 + c) * (size_t)TV
                    + (size_t)(2 * lhi) * VDIM + vc;             // [t][v]

  v8f acc = {};
#pragma unroll 4
  for (int kb = 0; kb < TDIM; kb += 4) {
    v2f a, b;
    a.x = attb[(size_t)kb * TDIM];          // A[m=q][k=t] = att[t][q]
    a.y = attb[(size_t)kb * TDIM + TDIM];
    b.x = gb[(size_t)kb * VDIM];
    b.y = gb[(size_t)kb * VDIM + VDIM];
    acc = wmma4(a, b, acc);
  }
#pragma unroll
  for (int rr = 0; rr < 8; ++rr) {
    int q = q0 + rr + 8 * lhi;
    if (vok)
      y2[(((size_t)n * 192 + s * 64 + c) * TDIM + q) * VDIM + v] = acc[rr];
  }
}

// ---------------------------------------------------------------------------
extern "C" void kernel_launch(void* const* d_in, const int* in_sizes, int n_in,
                              void* d_out, int out_size, void* d_ws, size_t ws_size,
                              hipStream_t stream)
{
  (void)in_sizes; (void)n_in; (void)out_size; (void)ws_size;

  const float* x      = (const float*)d_in[0];
  const float* Amat   = (const float*)d_in[1];
  const float* gcn_w  = (const float*)d_in[2];
  const float* gcn_b  = (const float*)d_in[3];
  const float* gcn_g  = (const float*)d_in[4];
  const float* gcn_bb = (const float*)d_in[5];
  const float* gcn_m  = (const float*)d_in[6];
  const float* gcn_v  = (const float*)d_in[7];
  const float* pe     = (const float*)d_in[8];
  const float* qkv_w  = (const float*)d_in[9];
  const float* qkv_b  = (const float*)d_in[10];
  const float* alphas = (const float*)d_in[11];
  const float* att1s  = (const float*)d_in[12];
  const float* on_w   = (const float*)d_in[13];
  const float* on_b   = (const float*)d_in[14];
  const float* on_g   = (const float*)d_in[15];
  const float* on_bb  = (const float*)d_in[16];
  const float* on_m   = (const float*)d_in[17];
  const float* on_v   = (const float*)d_in[18];
  const float* ff_w   = (const float*)d_in[19];
  const float* ff_b   = (const float*)d_in[20];
  const float* ff_g   = (const float*)d_in[21];
  const float* ff_bb  = (const float*)d_in[22];
  const float* ff_m   = (const float*)d_in[23];
  const float* ff_v   = (const float*)d_in[24];
  float* out = (float*)d_out;

  const int NB = 32;
  const size_t G = (size_t)NB * 64 * TV;        // one (N,64,T,V) tensor
  float* ws     = (float*)d_ws;
  float* gbuf   = ws;                           // g  (N,64,T,V)
  float* hbuf   = ws + G;                       // h, reused as y3
  float* qkbuf  = ws + 2 * G;                   // qk (N,96,T,V)
  float* y2buf  = ws + 2 * G;                   // y2 (N,192,T,V) reuses dead qk
  float* attbuf = ws + 2 * G + (size_t)NB * 192 * TV;  // att (N,3,64,64)

  auto nb = [](long tiles) { return (int)((tiles + 7) / 8); };

  // 1) h = gcn_w @ x + gcn_b
  long t1 = (long)NB * (64 / 16) * (TV / 16);
  conv1x1_wmma<0, 64><<<nb(t1), 256, 0, stream>>>(x, nullptr, gcn_w, gcn_b,
      nullptr, nullptr, nullptr, nullptr, nullptr, hbuf, 64, (int)t1);

  // 2) g = relu(bn(h @ A^T) + x)
  long t2 = (long)(NB * 64 * 64 / 16) * 13;
  amix_bn_relu_wmma<<<nb(t2), 256, 0, stream>>>(hbuf, Amat, x,
      gcn_g, gcn_bb, gcn_m, gcn_v, gbuf, (int)t2);

  // 3) qk = qkv_w @ (g + pe) + qkv_b
  long t3 = (long)NB * (96 / 16) * (TV / 16);
  conv1x1_wmma<1, 64><<<nb(t3), 256, 0, stream>>>(gbuf, pe, qkv_w, qkv_b,
      nullptr, nullptr, nullptr, nullptr, nullptr, qkbuf, 96, (int)t3);

  // 4) att = tanh(q.kT / (QK*V))*alpha + att1s
  long t4 = (long)NB * 3 * 16;
  attn_scores_wmma<<<nb(t4), 256, 0, stream>>>(qkbuf, alphas, att1s, attbuf,
                                               (int)t4);

  // 5) y2 = att^T @ g  (per n, head, channel)
  long t5 = (long)NB * 3 * 64 * 52;
  attn_apply_wmma<<<nb(t5), 256, 0, stream>>>(attbuf, gbuf, y2buf, (int)t5);

  // 6) y3 = leaky_relu(bn(on_w @ y2 + on_b) + g, 0.1)
  long t6 = (long)NB * (64 / 16) * (TV / 16);
  conv1x1_wmma<2, 192><<<nb(t6), 256, 0, stream>>>(y2buf, nullptr, on_w, on_b,
      on_g, on_bb, on_m, on_v, gbuf, hbuf, 64, (int)t6);

  // 7) out = leaky_relu(bn(ff_w @ y3 + ff_b) + g, 0.1)
  conv1x1_wmma<2, 64><<<nb(t6), 256, 0, stream>>>(hbuf, nullptr, ff_w, ff_b,
      ff_g, ff_bb, ff_m, ff_v, gbuf, out, 64, (int)t6);
}